// MyLoss_72258529788242
// MI455X (gfx1250) — compile-verified
//
#include <hip/hip_runtime.h>

// --- CDNA5 (gfx1250) wave32 WMMA types ---------------------------------------
typedef float v2f __attribute__((ext_vector_type(2)));
typedef float v8f __attribute__((ext_vector_type(8)));
typedef float f32x4 __attribute__((ext_vector_type(4)));
typedef int   i32x4 __attribute__((ext_vector_type(4)));

#define NBLOCKS 2048
#define NTHREADS 256

__device__ __forceinline__ void acc_elem(float xe, int te,
                                         float& pos, float& neg, float& cnt) {
    float tf = (float)te;                 // targets are strictly 0/1
    float v  = te ? xe : (1.0f - xe);     // pick the arg that matters
    float nl = -__logf(v);                // single v_log_f32 per element
    pos += tf * nl;
    neg += (1.0f - tf) * nl;
    cnt += tf;
}

// -----------------------------------------------------------------------------
// Kernel 1: streaming pass. Each thread accumulates (cnt, pos, neg) over its
// grid-stride share using NT B128 loads (2x unrolled for MLP); wave reduction
// via v_wmma_f32_16x16x4_f32; one (cnt,pos,neg) triple per block to workspace.
// -----------------------------------------------------------------------------
__global__ __launch_bounds__(NTHREADS)
void wbce_partial_kernel(const float* __restrict__ x,
                         const int*   __restrict__ t,
                         float* __restrict__ ws,
                         int nVec, long long n, int nb) {
    const int tid    = threadIdx.x;
    const int gtid   = (int)(blockIdx.x * NTHREADS) + tid;
    const int stride = (int)(gridDim.x * NTHREADS);

    float pos = 0.0f, neg = 0.0f, cnt = 0.0f;

    const f32x4* __restrict__ x4 = (const f32x4*)x;
    const i32x4* __restrict__ t4 = (const i32x4*)t;

    // Main vectorized stream: 16B NT loads (512MB stream > 192MB L2), 2x
    // unrolled so 4 B128 loads are in flight before any use.
    int i = gtid;
    const int nPair = nVec - stride;      // last index for which i+stride is valid
    for (; i < nPair; i += 2 * stride) {
        f32x4 xv0 = __builtin_nontemporal_load(&x4[i]);
        i32x4 tv0 = __builtin_nontemporal_load(&t4[i]);
        f32x4 xv1 = __builtin_nontemporal_load(&x4[i + stride]);
        i32x4 tv1 = __builtin_nontemporal_load(&t4[i + stride]);
#pragma unroll
        for (int k = 0; k < 4; ++k) acc_elem(xv0[k], tv0[k], pos, neg, cnt);
#pragma unroll
        for (int k = 0; k < 4; ++k) acc_elem(xv1[k], tv1[k], pos, neg, cnt);
    }
    if (i < nVec) {                       // leftover single chunk (uniform)
        f32x4 xv = __builtin_nontemporal_load(&x4[i]);
        i32x4 tv = __builtin_nontemporal_load(&t4[i]);
#pragma unroll
        for (int k = 0; k < 4; ++k) acc_elem(xv[k], tv[k], pos, neg, cnt);
    }
    // Scalar tail (empty for this problem size; kept for safety).
    for (long long j = (long long)nVec * 4 + gtid; j < n; j += stride) {
        acc_elem(x[j], t[j], pos, neg, cnt);
    }

    // ---- Wave32 reduction via WMMA (EXEC all-1s here; reconverged) ----------
    // A-matrix 16x4 f32: VGPR0 = {K0 (lanes0-15), K2 (lanes16-31)},
    //                    VGPR1 = {K1, K3}.  So A[m][0]=pos_m, A[m][1]=neg_m,
    //                    A[m][2]=pos_{m+16}, A[m][3]=neg_{m+16}.
    // B-matrix 4x16 f32 selector: cols 0-7 take K0+K2 (pos), cols 8-15 take
    //                    K1+K3 (neg). B VGPR0 = rows K0/K2, VGPR1 = rows K1/K3.
    const int lane = tid & 31;
    const float sel = ((lane & 15) < 8) ? 1.0f : 0.0f;
    v2f aPN; aPN[0] = pos; aPN[1] = neg;
    v2f aC;  aC[0]  = cnt; aC[1]  = 0.0f;
    v2f b;   b[0]   = sel; b[1]   = 1.0f - sel;
    v8f c0 = {};

    v8f d  = __builtin_amdgcn_wmma_f32_16x16x4_f32(
                 false, aPN, false, b, (short)0, c0, false, false);
    v8f dc = __builtin_amdgcn_wmma_f32_16x16x4_f32(
                 false, aC,  false, b, (short)0, c0, false, false);

    // Sum the 8 D VGPRs: lanes 0-7 hold sum(pos) over lanes 0-7,16-23;
    // lanes 16-23 hold sum(pos) over lanes 8-15,24-31; lanes 8-15 / 24-31
    // hold the corresponding neg halves. One xor-16 completes the wave sum.
    float s8 = d[0] + d[1] + d[2] + d[3] + d[4] + d[5] + d[6] + d[7];
    float c8 = dc[0] + dc[1] + dc[2] + dc[3] + dc[4] + dc[5] + dc[6] + dc[7];
    s8 += __shfl_xor(s8, 16, 32);
    c8 += __shfl_xor(c8, 16, 32);
    const float posW = __shfl(s8, 0, 32);   // full-wave sum of pos
    const float negW = __shfl(s8, 8, 32);   // full-wave sum of neg
    const float cntW = __shfl(c8, 0, 32);   // full-wave sum of cnt

    // ---- Cross-wave (8 waves) reduction through LDS -------------------------
    __shared__ float lds[8][3];
    const int wave = tid >> 5;
    if ((tid & 31) == 0) {
        lds[wave][0] = cntW;
        lds[wave][1] = posW;
        lds[wave][2] = negW;
    }
    __syncthreads();
    if (tid < 8) {
        float c = lds[tid][0];
        float p = lds[tid][1];
        float g = lds[tid][2];
#pragma unroll
        for (int off = 4; off >= 1; off >>= 1) {
            c += __shfl_xor(c, off, 32);
            p += __shfl_xor(p, off, 32);
            g += __shfl_xor(g, off, 32);
        }
        if (tid == 0) {
            ws[blockIdx.x]          = c;   // SoA: coalesced in the finisher
            ws[nb + blockIdx.x]     = p;
            ws[2 * nb + blockIdx.x] = g;
        }
    }
}

// -----------------------------------------------------------------------------
// Kernel 2: deterministic finisher. One block reduces the per-block triples
// and computes the weighted loss scalar.
// -----------------------------------------------------------------------------
__global__ __launch_bounds__(NTHREADS)
void wbce_final_kernel(const float* __restrict__ ws,
                       float* __restrict__ out,
                       int nb, float n_elems) {
    const int tid = threadIdx.x;
    float c = 0.0f, p = 0.0f, g = 0.0f;
    for (int i = tid; i < nb; i += NTHREADS) {
        c += ws[i];
        p += ws[nb + i];
        g += ws[2 * nb + i];
    }
#pragma unroll
    for (int off = 16; off >= 1; off >>= 1) {
        c += __shfl_xor(c, off, 32);
        p += __shfl_xor(p, off, 32);
        g += __shfl_xor(g, off, 32);
    }
    __shared__ float lds[8][3];
    const int wave = tid >> 5;
    if ((tid & 31) == 0) {
        lds[wave][0] = c; lds[wave][1] = p; lds[wave][2] = g;
    }
    __syncthreads();
    if (tid == 0) {
        float C = 0.0f, P = 0.0f, G = 0.0f;
#pragma unroll
        for (int w = 0; w < 8; ++w) {
            C += lds[w][0]; P += lds[w][1]; G += lds[w][2];
        }
        const float length = n_elems + 1.0f;         // matches f32 cast in ref
        const float wp = length / (C + 1.0f);
        const float wn = length / (length - C);
        out[0] = ((wp + 1.0f) * P + (wn + 1.0f) * G) / n_elems;
    }
}

// -----------------------------------------------------------------------------
extern "C" void kernel_launch(void* const* d_in, const int* in_sizes, int n_in,
                              void* d_out, int out_size, void* d_ws, size_t ws_size,
                              hipStream_t stream) {
    const float* x = (const float*)d_in[0];
    const int*   t = (const int*)d_in[1];
    float* out = (float*)d_out;
    float* ws  = (float*)d_ws;   // needs 3 * NBLOCKS floats = 24 KB

    const long long n    = (long long)in_sizes[0];
    const int       nVec = (int)(n >> 2);

    wbce_partial_kernel<<<NBLOCKS, NTHREADS, 0, stream>>>(x, t, ws, nVec, n, NBLOCKS);
    wbce_final_kernel<<<1, NTHREADS, 0, stream>>>(ws, out, NBLOCKS, (float)n);
}